// MatrixOut_29154238005852
// MI455X (gfx1250) — compile-verified
//
#include <hip/hip_runtime.h>
#include <hip/hip_bf16.h>

// ---------------------------------------------------------------------------
// Problem constants (match reference)
// ---------------------------------------------------------------------------
#define NN       160              // N_NODES
#define NE       25440            // 160*159 directed edges
#define ND       128              // NODE_DIM
#define FEAT     3136             // HID(64) * 49
#define O3D      1568             // BLK(32) * 49
#define NPW      1792             // NUM_PATH
#define NB       14               // NUM_BIAS
#define OD       14               // OUT_DIM
#define MAT      196              // 14*14
#define NPATH    19
#define W3J_TOT  1225

typedef float v2f __attribute__((ext_vector_type(2)));
typedef float v8f __attribute__((ext_vector_type(8)));

__device__ __forceinline__ v8f wmma4(v2f a, v2f b, v8f c) {
  // V_WMMA_F32_16X16X4_F32 : D = A(16x4) * B(4x16) + C
  return __builtin_amdgcn_wmma_f32_16x16x4_f32(false, a, false, b, (short)0, c,
                                               false, false);
}

// ---------------------------------------------------------------------------
// Static tables: irreps and tensor-product paths
// IRR_FEAT l = {0,1,1,2,2,3,3,4,4};  cumulative (2l+1): 0,1,4,7,12,17,24,31,40,49
// IRR_OUT  (mul,l) = (3,0),(2,1),(1,2)
// 19 expansion paths (i,j,k) derived from parity + triangle rule.
// ---------------------------------------------------------------------------
__constant__ int g_cumd[10] = {0, 1, 4, 7, 12, 17, 24, 31, 40, 49};
__constant__ int g_dirr[9]  = {1, 3, 3, 5, 5, 7, 7, 9, 9};

__constant__ int P_di[NPATH]   = {1,1,1,3,3,3,3,3,3,5,5,5,5,5,5,7,7,7,9};
__constant__ int P_xoff[NPATH] = {0,0,0,32,32,32,32,128,128,224,224,384,384,384,384,544,544,768,1280};
__constant__ int P_mj[NPATH]   = {3,2,1,3,2,2,1,2,1,2,1,3,2,1,1,2,1,1,1};
__constant__ int P_dj[NPATH]   = {1,3,5,1,3,3,5,3,5,3,5,1,3,5,5,3,5,5,5};
__constant__ int P_rowb[NPATH] = {0,3,9,0,3,3,9,3,9,3,9,0,3,9,9,3,9,9,9};
__constant__ int P_mk[NPATH]   = {3,2,1,2,3,1,2,2,1,1,2,1,2,3,1,1,2,1,1};
__constant__ int P_dk[NPATH]   = {1,3,5,3,1,5,3,3,5,5,3,5,3,1,5,5,3,5,5};
__constant__ int P_colb[NPATH] = {0,3,9,3,0,9,3,3,9,9,3,9,3,0,9,9,3,9,9};
__constant__ int P_wb[NPATH]   = {0,288,416,448,640,832,896,960,1088,1120,1184,1248,1344,1472,1568,1600,1664,1728,1760};
__constant__ int P_bb[NPATH]   = {0,9,13,-1,-1,-1,-1,-1,-1,-1,-1,-1,-1,-1,-1,-1,-1,-1,-1};
__constant__ int P_c3[NPATH]   = {0,1,10,35,44,53,98,143,170,245,320,395,420,465,490,615,720,825,1000};

// ---------------------------------------------------------------------------
// Wigner 3j (e3nn real basis) computed once on device in double precision.
// ---------------------------------------------------------------------------
struct cplx { double x, y; };
__device__ __forceinline__ cplx cmul(cplx a, cplx b) {
  return {a.x * b.x - a.y * b.y, a.x * b.y + a.y * b.x};
}
__device__ double dfact(int n) {
  double r = 1.0;
  for (int i = 2; i <= n; ++i) r *= (double)i;
  return r;
}
__device__ double su2_cg(int j1, int m1, int j2, int m2, int j3, int m3) {
  int vmin = -j1 + j2 + m3; if (-j1 + m1 > vmin) vmin = -j1 + m1; if (0 > vmin) vmin = 0;
  int vmax = j2 + j3 + m1; if (j3 - j1 + j2 < vmax) vmax = j3 - j1 + j2; if (j3 + m3 < vmax) vmax = j3 + m3;
  double c = sqrt((double)(2 * j3 + 1) * dfact(j3 + j1 - j2) * dfact(j3 - j1 + j2) *
                  dfact(j1 + j2 - j3) * dfact(j3 + m3) * dfact(j3 - m3) /
                  (dfact(j1 + j2 + j3 + 1) * dfact(j1 - m1) * dfact(j1 + m1) *
                   dfact(j2 - m2) * dfact(j2 + m2)));
  double s = 0.0;
  for (int v = vmin; v <= vmax; ++v) {
    double sg = ((v + j2 + m2) & 1) ? -1.0 : 1.0;
    s += sg * dfact(j2 + j3 + m1 - v) * dfact(j1 - m1 + v) /
         (dfact(v) * dfact(j3 - j1 + j2 - v) * dfact(j3 + m3 - v) * dfact(v + j1 - j2 - m3));
  }
  return c * s;
}
__device__ void build_q(int l, cplx* q) {
  int D = 2 * l + 1;
  for (int i = 0; i < D * D; ++i) q[i] = {0.0, 0.0};
  const double r2 = 0.70710678118654752440;
  for (int m = -l; m < 0; ++m) {
    q[(l + m) * D + (l - m)] = {r2, 0.0};
    q[(l + m) * D + (l + m)] = {0.0, -r2};
  }
  q[l * D + l] = {1.0, 0.0};
  for (int m = 1; m <= l; ++m) {
    double sg = (m & 1) ? -1.0 : 1.0;
    q[(l + m) * D + (l + m)] = {sg * r2, 0.0};
    q[(l + m) * D + (l - m)] = {0.0, sg * r2};
  }
  cplx f;
  switch (l & 3) {  // (-i)^l
    case 0: f = {1.0, 0.0}; break;
    case 1: f = {0.0, -1.0}; break;
    case 2: f = {-1.0, 0.0}; break;
    default: f = {0.0, 1.0}; break;
  }
  for (int i = 0; i < D * D; ++i) q[i] = cmul(q[i], f);
}

__global__ void mo_w3j_init(float* __restrict__ C3out) {
  int p = threadIdx.x;
  if (p >= NPATH) return;
  int l1 = (P_dj[p] - 1) / 2, l2 = (P_dk[p] - 1) / 2, l3 = (P_di[p] - 1) / 2;
  int D1 = 2 * l1 + 1, D2 = 2 * l2 + 1, D3 = 2 * l3 + 1;
  double C[225];
  for (int i = 0; i < D1 * D2 * D3; ++i) C[i] = 0.0;
  double nrm = 1.0 / sqrt((double)(2 * l3 + 1));
  for (int m1 = -l1; m1 <= l1; ++m1)
    for (int m2 = -l2; m2 <= l2; ++m2) {
      int m3 = m1 + m2;
      if (m3 < -l3 || m3 > l3) continue;
      C[((l1 + m1) * D2 + (l2 + m2)) * D3 + (l3 + m3)] =
          su2_cg(l1, m1, l2, m2, l3, m3) * nrm;
    }
  cplx q1[25], q2[25], q3[81];
  build_q(l1, q1); build_q(l2, q2); build_q(l3, q3);
  float* out = C3out + P_c3[p];
  for (int jj = 0; jj < D1; ++jj)
    for (int ll = 0; ll < D2; ++ll)
      for (int mm = 0; mm < D3; ++mm) {
        double accx = 0.0;
        for (int i = 0; i < D1; ++i)
          for (int k = 0; k < D2; ++k) {
            cplx a = cmul(q1[i * D1 + jj], q2[k * D2 + ll]);
            for (int n = 0; n < D3; ++n) {
              double cv = C[(i * D2 + k) * D3 + n];
              if (cv == 0.0) continue;
              cplx b = {q3[n * D3 + mm].x, -q3[n * D3 + mm].y};  // conj
              cplx t = cmul(a, b);
              accx += t.x * cv;
            }
          }
        out[(jj * D2 + ll) * D3 + mm] = (float)accx;
      }
}

// ---------------------------------------------------------------------------
// GEMM 1: Hidden = silu(X @ W1 + b1).  X gathered:
//   mode 0 (nodes): X[r] = embed[r]            (K = 128)
//   mode 1 (edges): X[r] = [embed[src],embed[dst]] (K = 256), src/dst analytic
// One wave computes a 16 x (16*NT) tile: the A fragment is loaded once per
// K-step and reused across NT independent WMMA accumulator chains.
// ---------------------------------------------------------------------------
template <int NT>
__global__ __launch_bounds__(32) void mo_hidden_gemm(
    const float* __restrict__ embed, const float* __restrict__ w1,
    const float* __restrict__ b1, float* __restrict__ H, int M, int K,
    int edge_base, int edges) {
  const int lane = threadIdx.x & 31;
  const int mrow = lane & 15;
  const int kb = (lane >> 4) << 1;                    // 0 or 2
  const int ncol0 = (blockIdx.y * NT * 16) + mrow;    // first col of this wave

  int row = blockIdx.x * 16 + mrow;
  int arow = row < M ? row : M - 1;
  const float *pa0, *pa1;
  if (edges) {
    int e = edge_base + arow;
    int s = e / (NN - 1);
    int t = e % (NN - 1);
    int d = t + (t >= s ? 1 : 0);
    pa0 = embed + (size_t)s * ND;
    pa1 = embed + (size_t)d * ND;
  } else {
    pa0 = embed + (size_t)arow * ND;
    pa1 = pa0;
  }

  v8f c[NT];
#pragma unroll
  for (int t = 0; t < NT; ++t) c[t] = (v8f){};

  for (int k0 = 0; k0 < K; k0 += 4) {
    int k = k0 + kb;
    v2f a;
    if (k < ND) { a.x = pa0[k];      a.y = pa0[k + 1]; }
    else        { a.x = pa1[k - ND]; a.y = pa1[k - ND + 1]; }
    if (k0 + 16 < K) {
      int kn = k0 + 16 + kb;
      __builtin_prefetch(kn < ND ? (pa0 + kn) : (pa1 + kn - ND), 0, 1);
    }
    const float* pb = w1 + (size_t)k * ND + ncol0;
#pragma unroll
    for (int t = 0; t < NT; ++t) {
      v2f b;
      b.x = pb[t * 16];
      b.y = pb[ND + t * 16];
      c[t] = wmma4(a, b, c[t]);
    }
  }
  int rbase = blockIdx.x * 16 + ((lane >> 4) << 3);
#pragma unroll
  for (int t = 0; t < NT; ++t) {
    int col = ncol0 + t * 16;
    float bias = b1[col];
#pragma unroll
    for (int r = 0; r < 8; ++r) {
      int rr = rbase + r;
      if (rr < M) {
        float x = c[t][r] + bias;
        H[(size_t)rr * ND + col] = x / (1.0f + __expf(-x));  // silu
      }
    }
  }
}

// GEMM 2: Y = H(Mx128) @ W2(128 x NOUT) + b2.  16 x (16*NT) per wave.
template <int NT>
__global__ __launch_bounds__(32) void mo_out_gemm(
    const float* __restrict__ H, const float* __restrict__ w2,
    const float* __restrict__ b2, float* __restrict__ Y, int M, int NOUT) {
  const int lane = threadIdx.x & 31;
  const int mrow = lane & 15;
  const int kb = (lane >> 4) << 1;
  const int col0 = (blockIdx.y * NT * 16) + mrow;

  int row = blockIdx.x * 16 + mrow;
  int arow = row < M ? row : M - 1;
  const float* pa = H + (size_t)arow * ND;

  v8f c[NT];
#pragma unroll
  for (int t = 0; t < NT; ++t) c[t] = (v8f){};

  for (int k0 = 0; k0 < ND; k0 += 4) {
    int k = k0 + kb;
    v2f a;
    a.x = pa[k]; a.y = pa[k + 1];
    if (k0 + 16 < ND) __builtin_prefetch(pa + k0 + 16 + kb, 0, 1);
    const float* pb = w2 + (size_t)k * NOUT + col0;
#pragma unroll
    for (int t = 0; t < NT; ++t) {
      v2f b;
      bool ok = (col0 + t * 16) < NOUT;
      b.x = ok ? pb[t * 16] : 0.0f;
      b.y = ok ? pb[(size_t)NOUT + t * 16] : 0.0f;
      c[t] = wmma4(a, b, c[t]);
    }
  }
  int rbase = blockIdx.x * 16 + ((lane >> 4) << 3);
#pragma unroll
  for (int t = 0; t < NT; ++t) {
    int col = col0 + t * 16;
    if (col < NOUT) {
      float bias = b2[col];
#pragma unroll
      for (int r = 0; r < 8; ++r) {
        int rr = rbase + r;
        if (rr < M) Y[(size_t)rr * NOUT + col] = c[t][r] + bias;
      }
    }
  }
}

// GEMM 3: o3_linear. blockIdx.y = comp in [0,49) -> (irrep s, m-index).
// Per component: Y(:, 32) = X(:,64) @ W[s] * (1/8); one wave does both 16-col
// halves of the 32-wide output, reusing the strided A fragment.
__global__ __launch_bounds__(32) void mo_o3lin_gemm(
    const float* __restrict__ feat, const float* __restrict__ W,
    float* __restrict__ Y, int M) {
  const int lane = threadIdx.x & 31;
  const int mrow = lane & 15;
  const int kb = (lane >> 4) << 1;
  const int comp = blockIdx.y;
  int s = 0;
  while (g_cumd[s + 1] <= comp) ++s;
  const int d = g_dirr[s];
  const int cloc = comp - g_cumd[s];
  const int xbase = 64 * g_cumd[s] + cloc;
  const int ybase = 32 * g_cumd[s] + cloc;

  int row = blockIdx.x * 16 + mrow;
  int arow = row < M ? row : M - 1;
  const float* pa = feat + (size_t)arow * FEAT + xbase;
  const float* pw = W + s * 64 * 32;

  v8f c0 = {}, c1 = {};
  for (int k0 = 0; k0 < 64; k0 += 4) {
    int k = k0 + kb;
    v2f a, b0, b1v;
    a.x = pa[(size_t)k * d];
    a.y = pa[(size_t)(k + 1) * d];
    b0.x = pw[k * 32 + mrow];
    b0.y = pw[(k + 1) * 32 + mrow];
    b1v.x = pw[k * 32 + 16 + mrow];
    b1v.y = pw[(k + 1) * 32 + 16 + mrow];
    c0 = wmma4(a, b0, c0);
    c1 = wmma4(a, b1v, c1);
  }
  int rbase = blockIdx.x * 16 + ((lane >> 4) << 3);
#pragma unroll
  for (int r = 0; r < 8; ++r) {
    int rr = rbase + r;
    if (rr < M) {
      float* yp = Y + (size_t)rr * O3D + ybase;
      yp[(size_t)mrow * d] = 0.125f * c0[r];
      yp[(size_t)(mrow + 16) * d] = 0.125f * c1[r];
    }
  }
}

// ---------------------------------------------------------------------------
// Expansion: one wave per sample, 8 samples per 256-thread block.
// r[u,v,k] = sum_w W[wb + (w*mj+u)*mk+v] * X[xoff + w*di + k]  (+bias for l=0)
// out[rowb+u*dj+i, colb+v*dk+j] += (1/32) sum_k C3[(i*dk+j)*di+k] * r[u,v,k]
// ---------------------------------------------------------------------------
__global__ __launch_bounds__(256) void mo_expansion(
    const float* __restrict__ X, const float* __restrict__ Wt,
    const float* __restrict__ Bs, const float* __restrict__ C3,
    float* __restrict__ Pre, int cnt) {
  __shared__ float rbuf[8][32];
  __shared__ float obuf[8][MAT];
  const int wave = threadIdx.x >> 5;
  const int lane = threadIdx.x & 31;
  int samp = blockIdx.x * 8 + wave;
  int asamp = samp < cnt ? samp : cnt - 1;
  const float* x = X + (size_t)asamp * O3D;
  const float* w = Wt + (size_t)asamp * NPW;
  const float* bb = Bs + (size_t)asamp * NB;

  for (int t = lane; t < MAT; t += 32) obuf[wave][t] = 0.0f;
  __syncthreads();

#pragma unroll 1
  for (int p = 0; p < NPATH; ++p) {
    const int di = P_di[p], mj = P_mj[p], mk = P_mk[p];
    const int dj = P_dj[p], dk = P_dk[p];
    const int nr = mj * mk * di;  // <= 20
    float r = 0.0f;
    if (lane < nr) {
      int k = lane % di;
      int uv = lane / di;
      int u = uv / mk, v = uv % mk;
      const float* wp = w + P_wb[p] + u * mk + v;
      const float* xp = x + P_xoff[p] + k;
      const int ws2 = mj * mk;
      for (int ww = 0; ww < 32; ++ww) r += wp[ww * ws2] * xp[ww * di];
      if (P_bb[p] >= 0 && k == 0) r += bb[P_bb[p] + uv];
    }
    rbuf[wave][lane] = r;
    __syncthreads();
    const int ncolsB = mk * dk;
    const int nout = mj * dj * ncolsB;  // <= 36
    const float* c3 = C3 + P_c3[p];
    for (int el = lane; el < nout; el += 32) {
      int rr = el / ncolsB, cc = el % ncolsB;
      int u = rr / dj, ii = rr % dj;
      int v = cc / dk, jj = cc % dk;
      float acc = 0.0f;
      const float* rp = &rbuf[wave][(u * mk + v) * di];
      const float* cp = c3 + (ii * dk + jj) * di;
      for (int k = 0; k < di; ++k) acc += cp[k] * rp[k];
      obuf[wave][(P_rowb[p] + rr) * OD + P_colb[p] + cc] += acc * 0.03125f;
    }
    __syncthreads();
  }
  if (samp < cnt) {
    float* o = Pre + (size_t)samp * MAT;
    for (int t = lane; t < MAT; t += 32) o[t] = obuf[wave][t];
  }
}

// ---------------------------------------------------------------------------
// Symmetrization. Off-diagonal uses closed-form stable argsort(dst) on the
// full directed graph: position e in dst-sorted order is edge (s', g) with
// g = e/159, s' = t + (t>=g), t = e%159; its original index is
// q = s'*159 + (g < s' ? g : g-1).
// ---------------------------------------------------------------------------
__global__ void mo_sym_diag(const float* __restrict__ Pre,
                            float* __restrict__ Out, int n) {
  int idx = blockIdx.x * blockDim.x + threadIdx.x;
  if (idx >= n * MAT) return;
  int s = idx / MAT, t = idx % MAT;
  int r = t / OD, c = t % OD;
  Out[idx] = 0.5f * (Pre[(size_t)s * MAT + r * OD + c] +
                     Pre[(size_t)s * MAT + c * OD + r]);
}

__global__ void mo_sym_off(const float* __restrict__ Pre,
                           float* __restrict__ Out) {
  int idx = blockIdx.x * blockDim.x + threadIdx.x;
  if (idx >= NE * MAT) return;
  int e = idx / MAT, t = idx % MAT;
  int r = t / OD, c = t % OD;
  int g = e / (NN - 1);
  int tt = e % (NN - 1);
  int sp = tt + (tt >= g ? 1 : 0);
  int q = sp * (NN - 1) + (g < sp ? g : g - 1);
  Out[idx] = 0.5f * (Pre[(size_t)e * MAT + r * OD + c] +
                     Pre[(size_t)q * MAT + c * OD + r]);
}

// ---------------------------------------------------------------------------
// Host launcher
// ---------------------------------------------------------------------------
extern "C" void kernel_launch(void* const* d_in, const int* in_sizes, int n_in,
                              void* d_out, int out_size, void* d_ws,
                              size_t ws_size, hipStream_t stream) {
  (void)in_sizes; (void)n_in; (void)out_size;
  const float* fii      = (const float*)d_in[0];
  const float* fij      = (const float*)d_in[1];
  const float* embed    = (const float*)d_in[2];
  /* d_in[3] = edge_index_full: permutation derived analytically */
  const float* wii      = (const float*)d_in[4];
  const float* wij      = (const float*)d_in[5];
  const float* ii_fc_w1 = (const float*)d_in[6];
  const float* ii_fc_b1 = (const float*)d_in[7];
  const float* ii_fc_w2 = (const float*)d_in[8];
  const float* ii_fc_b2 = (const float*)d_in[9];
  const float* ii_fb_w1 = (const float*)d_in[10];
  const float* ii_fb_b1 = (const float*)d_in[11];
  const float* ii_fb_w2 = (const float*)d_in[12];
  const float* ii_fb_b2 = (const float*)d_in[13];
  const float* ij_fc_w1 = (const float*)d_in[14];
  const float* ij_fc_b1 = (const float*)d_in[15];
  const float* ij_fc_w2 = (const float*)d_in[16];
  const float* ij_fc_b2 = (const float*)d_in[17];
  const float* ij_fb_w1 = (const float*)d_in[18];
  const float* ij_fb_b1 = (const float*)d_in[19];
  const float* ij_fb_w2 = (const float*)d_in[20];
  const float* ij_fb_b2 = (const float*)d_in[21];

  // Workspace layout (floats)
  const size_t F_C3 = 2048;
  const size_t F_PREOFF = (size_t)NE * MAT;
  const size_t F_PREDIAG = (size_t)NN * MAT;
  const size_t F_HN = (size_t)NN * ND;
  const size_t F_WTN = (size_t)NN * NPW;
  const size_t F_BSN = (size_t)NN * NB;
  const size_t F_XON = (size_t)NN * O3D;
  size_t fixedF = F_C3 + F_PREOFF + F_PREDIAG + F_HN + F_WTN + F_BSN + F_XON;
  size_t totalF = ws_size / sizeof(float);
  size_t avail = totalF > fixedF ? totalF - fixedF : 0;
  const long perEdge = ND + NPW + NB + O3D;  // 3502 floats per edge of chunk
  long C = (long)(avail / (size_t)perEdge);
  C = (C / 16) * 16;
  if (C < 16) C = 16;
  if (C > NE) C = NE;

  float* wsf     = (float*)d_ws;
  float* C3      = wsf;
  float* PreOff  = C3 + F_C3;
  float* PreDiag = PreOff + F_PREOFF;
  float* Hn      = PreDiag + F_PREDIAG;
  float* Wtn     = Hn + F_HN;
  float* Bsn     = Wtn + F_WTN;
  float* Xon     = Bsn + F_BSN;
  float* Hc      = Xon + F_XON;
  float* Wtc     = Hc + (size_t)C * ND;
  float* Bsc     = Wtc + (size_t)C * NPW;
  float* Xoc     = Bsc + (size_t)C * NB;

  float* out_diag = (float*)d_out;
  float* out_off  = out_diag + (size_t)NN * MAT;

  // Wigner 3j constants
  mo_w3j_init<<<1, 32, 0, stream>>>(C3);

  // ---- Node (diagonal) pipeline ----
  {
    int M = NN;
    dim3 gh((M + 15) / 16, 2);                       // NT=4 -> 128/64 = 2
    mo_hidden_gemm<4><<<gh, 32, 0, stream>>>(embed, ii_fc_w1, ii_fc_b1, Hn, M, ND, 0, 0);
    dim3 gw((M + 15) / 16, NPW / 64);                // NT=4 -> 28 col groups
    mo_out_gemm<4><<<gw, 32, 0, stream>>>(Hn, ii_fc_w2, ii_fc_b2, Wtn, M, NPW);
    mo_hidden_gemm<4><<<gh, 32, 0, stream>>>(embed, ii_fb_w1, ii_fb_b1, Hn, M, ND, 0, 0);
    dim3 gb((M + 15) / 16, 1);                       // NT=1, 14 cols guarded
    mo_out_gemm<1><<<gb, 32, 0, stream>>>(Hn, ii_fb_w2, ii_fb_b2, Bsn, M, NB);
    dim3 go((M + 15) / 16, 49);
    mo_o3lin_gemm<<<go, 32, 0, stream>>>(fii, wii, Xon, M);
    mo_expansion<<<(M + 7) / 8, 256, 0, stream>>>(Xon, Wtn, Bsn, C3, PreDiag, M);
  }

  // ---- Edge (off-diagonal) pipeline, chunked by workspace budget ----
  for (long base = 0; base < NE; base += C) {
    int cnt = (int)((NE - base) < C ? (NE - base) : C);
    dim3 gh((cnt + 15) / 16, 2);
    mo_hidden_gemm<4><<<gh, 32, 0, stream>>>(embed, ij_fc_w1, ij_fc_b1, Hc, cnt,
                                             2 * ND, (int)base, 1);
    dim3 gw((cnt + 15) / 16, NPW / 64);
    mo_out_gemm<4><<<gw, 32, 0, stream>>>(Hc, ij_fc_w2, ij_fc_b2, Wtc, cnt, NPW);
    mo_hidden_gemm<4><<<gh, 32, 0, stream>>>(embed, ij_fb_w1, ij_fb_b1, Hc, cnt,
                                             2 * ND, (int)base, 1);
    dim3 gb((cnt + 15) / 16, 1);
    mo_out_gemm<1><<<gb, 32, 0, stream>>>(Hc, ij_fb_w2, ij_fb_b2, Bsc, cnt, NB);
    dim3 go((cnt + 15) / 16, 49);
    mo_o3lin_gemm<<<go, 32, 0, stream>>>(fij + (size_t)base * FEAT, wij, Xoc, cnt);
    mo_expansion<<<(cnt + 7) / 8, 256, 0, stream>>>(Xoc, Wtc, Bsc, C3,
                                                    PreOff + (size_t)base * MAT,
                                                    cnt);
  }

  // ---- Symmetrize ----
  {
    int tot = NN * MAT;
    mo_sym_diag<<<(tot + 255) / 256, 256, 0, stream>>>(PreDiag, out_diag, NN);
    long tot2 = (long)NE * MAT;
    mo_sym_off<<<(int)((tot2 + 255) / 256), 256, 0, stream>>>(PreOff, out_off);
  }
}